// Acoustic_Simulation_45500883533977
// MI455X (gfx1250) — compile-verified
//
#include <hip/hip_runtime.h>

// Acoustic 2-D FDTD on gfx1250: per-step fused kernel, Laplacian via
// V_WMMA_F32_16X16X4_F32 (D = T*P + P*T on 16x16 tiles, T tridiagonal [1,-2,1]).

#define NXP 356
#define NYP 356
#define NT 300
#define SRC_N 2
#define RCV_N 200
#define TILES_X 23  // ceil(356/16)
#define TILES_Y 23
#define DT_F 1.0e-3f
#define INV_DX2 0.01f      // 1/(10*10)
#define SHS 21             // LDS row stride (odd -> conflict-free column reads)

typedef __attribute__((ext_vector_type(2))) float v2f;
typedef __attribute__((ext_vector_type(8))) float v8f;

__global__ __launch_bounds__(32) void fd_step_wmma(
    const float* __restrict__ vel, const float* __restrict__ damp,
    const float* __restrict__ stf, const int* __restrict__ src_x,
    const int* __restrict__ src_y, const int* __restrict__ rcv_x,
    const int* __restrict__ rcv_y, float* __restrict__ pbase,
    int i0, int i1, int i2, float* __restrict__ csg, int t)
{
  const int lane  = threadIdx.x;     // one wave32 per block, one 16x16 tile
  const int laneN = lane & 15;       // N (x within tile) for B/C/D layouts
  const int hi    = lane >> 4;       // lane half
  const int tile  = blockIdx.x;
  const int ty = tile / TILES_X, tx = tile % TILES_X;
  const int y0 = ty * 16, x0 = tx * 16;
  const int shot = blockIdx.z;
  const size_t NN = (size_t)NXP * NYP;
  const float* p0 = pbase + ((size_t)shot * 3 + i0) * NN;
  const float* p1 = pbase + ((size_t)shot * 3 + i1) * NN;
  float*       p2 = pbase + ((size_t)shot * 3 + i2) * NN;

  // Stage 18x18 halo tile of p1 into LDS; out-of-grid = 0 (Dirichlet pad,
  // matches the reference's jnp.pad zero boundary).
  __shared__ float sh[18 * SHS];
  for (int idx = lane; idx < 18 * 18; idx += 32) {
    int r = idx / 18, c = idx % 18;
    int gy = y0 + r - 1, gx = x0 + c - 1;
    float val = 0.0f;
    if (gy >= 0 && gy < NYP && gx >= 0 && gx < NXP)
      val = p1[(size_t)gy * NXP + gx];
    sh[r * SHS + c] = val;
  }
  __syncthreads();

  // acc = T*P + P*T  (tridiagonal T applied on both sides = 5-pt stencil core)
  v8f acc = {};
  #pragma unroll
  for (int kc = 0; kc < 4; ++kc) {
    const int K0 = kc * 4 + hi * 2;      // K indices this lane carries
    const int K1 = K0 + 1;
    // T chunk: identical per-lane values in A-layout (M=laneN,K) and
    // B-layout (K,N=laneN) because T is symmetric and both layouts use the
    // same (vgpr,lane-half) -> K striping.
    v2f tc;
    tc.x = (laneN == K0) ? -2.0f : (((laneN - K0 == 1) || (K0 - laneN == 1)) ? 1.0f : 0.0f);
    tc.y = (laneN == K1) ? -2.0f : (((laneN - K1 == 1) || (K1 - laneN == 1)) ? 1.0f : 0.0f);
    // P as B chunk: element P[K][N]  (column read, conflict-free via SHS=21)
    v2f pb;
    pb.x = sh[(K0 + 1) * SHS + laneN + 1];
    pb.y = sh[(K1 + 1) * SHS + laneN + 1];
    // P as A chunk: element P[M][K], M = laneN (row read)
    v2f pa;
    pa.x = sh[(laneN + 1) * SHS + K0 + 1];
    pa.y = sh[(laneN + 1) * SHS + K1 + 1];
    // acc += T*P
    acc = __builtin_amdgcn_wmma_f32_16x16x4_f32(false, tc, false, pb,
                                                (short)0, acc, false, false);
    // acc += P*T
    acc = __builtin_amdgcn_wmma_f32_16x16x4_f32(false, pa, false, tc,
                                                (short)0, acc, false, false);
  }

  // Source terms for this shot/step.
  const int sx = src_x[shot], sy = src_y[shot];
  const float sval = stf[shot * NT + t];
  const float vsrc = vel[(size_t)sy * NXP + sx];
  const float add_src = sval * (vsrc * DT_F) * (vsrc * DT_F) * INV_DX2; // s*alpha/(DX*DY)

  // Pointwise leapfrog + sponge update; C/D layout: vgpr v holds row M=v+8*hi.
  float p2o[8];
  #pragma unroll
  for (int v8 = 0; v8 < 8; ++v8) {
    const int M = v8 + hi * 8;
    const int gy = y0 + M, gx = x0 + laneN;
    float d = acc[v8];
    // Halo corrections for tile-boundary rows/cols of T*P + P*T.
    if (M == 0)      d += sh[0 * SHS + laneN + 1];
    if (M == 15)     d += sh[17 * SHS + laneN + 1];
    if (laneN == 0)  d += sh[(M + 1) * SHS + 0];
    if (laneN == 15) d += sh[(M + 1) * SHS + 17];
    const float lap = d * INV_DX2;
    float out = 0.0f;
    if (gy < NYP && gx < NXP) {
      const size_t g = (size_t)gy * NXP + gx;
      const float p0v = p0[g];
      const float vv  = vel[g];
      const float dmp = damp[g] * DT_F;
      const float alpha = (vv * DT_F) * (vv * DT_F);
      const float p1v = sh[(M + 1) * SHS + laneN + 1];
      out = (2.0f * p1v - (1.0f - dmp) * p0v + alpha * lap) / (1.0f + dmp);
      if (gy == sy && gx == sx) out += add_src;  // point source injection
      p2[g] = out;
    }
    p2o[v8] = out;
  }

  // Receiver gather: each tile records only its own freshly computed cells
  // (no cross-block dependency; receiver row != source row in the setup,
  // and even coincidence would be ordered correctly since out includes src).
  for (int r = 0; r < RCV_N; ++r) {
    const int ry = rcv_y[r] - y0;
    const int rx = rcv_x[r] - x0;
    if ((unsigned)ry < 16u && (unsigned)rx < 16u) {
      if (laneN == rx && hi == (ry >> 3)) {
        const int rsel = ry & 7;
        float rv = p2o[0];
        #pragma unroll
        for (int q = 1; q < 8; ++q) rv = (rsel == q) ? p2o[q] : rv;
        csg[((size_t)shot * NT + t) * RCV_N + r] = rv;
      }
    }
  }
}

__global__ void zero_fields(float* __restrict__ p, size_t n) {
  size_t i = (size_t)blockIdx.x * blockDim.x + threadIdx.x;
  if (i < n) p[i] = 0.0f;
}

extern "C" void kernel_launch(void* const* d_in, const int* in_sizes, int n_in,
                              void* d_out, int out_size, void* d_ws, size_t ws_size,
                              hipStream_t stream) {
  (void)in_sizes; (void)n_in; (void)out_size; (void)ws_size;
  const float* vel  = (const float*)d_in[0];   // [356,356]
  const float* damp = (const float*)d_in[1];   // [356,356]
  const float* stf  = (const float*)d_in[2];   // [2,300]
  const int*   sx   = (const int*)d_in[3];     // [2]
  const int*   sy   = (const int*)d_in[4];     // [2]
  const int*   rx   = (const int*)d_in[5];     // [200]
  const int*   ry   = (const int*)d_in[6];     // [200]

  float* out  = (float*)d_out;
  float* csg  = out;                                   // [2,300,200]
  float* forw = out + (size_t)SRC_N * NT * RCV_N;      // [2,356,356]
  float* pbase = (float*)d_ws;                         // 2 shots x 3 fields
  const size_t NN = (size_t)NXP * NYP;

  // Zero the rotating wavefield buffers (harness poisons d_ws).
  const size_t tot = (size_t)SRC_N * 3 * NN;
  zero_fields<<<dim3((unsigned)((tot + 255) / 256)), dim3(256), 0, stream>>>(pbase, tot);

  int i0 = 0, i1 = 1, i2 = 2;
  dim3 grid(TILES_X * TILES_Y, 1, SRC_N);
  dim3 block(32);
  for (int t = 0; t < NT; ++t) {
    fd_step_wmma<<<grid, block, 0, stream>>>(vel, damp, stf, sx, sy, rx, ry,
                                             pbase, i0, i1, i2, csg, t);
    const int n0 = i1, n1 = i2, n2 = i0;   // (p0,p1) <- (p1,p2)
    i0 = n0; i1 = n1; i2 = n2;
  }
  // After the final rotation, buffer i1 holds the last p2 (== reference pf).
  for (int s = 0; s < SRC_N; ++s) {
    hipMemcpyAsync(forw + (size_t)s * NN,
                   pbase + ((size_t)s * 3 + i1) * NN,
                   NN * sizeof(float), hipMemcpyDeviceToDevice, stream);
  }
}